// MultiHeadDistanceLayer_7060926234758
// MI455X (gfx1250) — compile-verified
//
#include <hip/hip_runtime.h>

typedef float v2f __attribute__((ext_vector_type(2)));
typedef float v8f __attribute__((ext_vector_type(8)));
typedef unsigned int u32x4 __attribute__((ext_vector_type(4)));
typedef int i32x4 __attribute__((ext_vector_type(4)));
typedef int i32x8 __attribute__((ext_vector_type(8)));

#define B_BATCH 4
#define L_SEQ   2048
#define D_IN    256
#define H_HEADS 8
#define KD      64
#define NCOLS   512          // H*KD
#define TBL_N   2176         // 2048 + 128 prior-table entries per block
#define NT      (L_SEQ / 16) // key tiles

#if __has_builtin(__builtin_amdgcn_tensor_load_to_lds) && \
    __has_builtin(__builtin_amdgcn_s_wait_tensorcnt)
#define USE_TDM 1
#else
#define USE_TDM 0
#endif

// ---------------------------------------------------------------------------
// Projection: (B*L, 256) x (256, 512) + per-head bias  ->  q / k workspace
// One 16x16 output tile per wave; 64 x V_WMMA_F32_16X16X4_F32 over D=256.
// ---------------------------------------------------------------------------
__global__ __launch_bounds__(128)
void proj_kernel(const float* __restrict__ x,
                 const float* __restrict__ Wq, const float* __restrict__ bq,
                 const float* __restrict__ Wk, const float* __restrict__ bk,
                 float* __restrict__ qout, float* __restrict__ kout)
{
    const int wave = threadIdx.x >> 5;
    const int lane = threadIdx.x & 31;
    const int half = lane >> 4;
    const int ln   = lane & 15;

    const int rbase = (blockIdx.x * 4 + wave) * 16;   // row tile in B*L
    const int nbase = blockIdx.y * 16;                // col tile in 512
    const float* W    = blockIdx.z ? Wk : Wq;
    const float* bias = blockIdx.z ? bk : bq;
    float*       outp = blockIdx.z ? kout : qout;

    const float* arow = x + (size_t)(rbase + ln) * D_IN + 2 * half;
    const float* bcol = W + nbase + ln;

    v8f acc = {};
#pragma unroll
    for (int kk = 0; kk < D_IN / 4; ++kk) {
        const int kd = kk * 4;
        v2f a = *(const v2f*)(arow + kd);
        v2f bb;
        bb.x = bcol[(size_t)(kd + 2 * half)     * NCOLS];
        bb.y = bcol[(size_t)(kd + 2 * half + 1) * NCOLS];
        acc = __builtin_amdgcn_wmma_f32_16x16x4_f32(false, a, false, bb,
                                                    (short)0, acc, false, false);
    }

    const int   n  = nbase + ln;
    const float bv = bias[n >> 6];                    // per-head scalar bias
#pragma unroll
    for (int j = 0; j < 8; ++j) {
        const int r = rbase + j + 8 * half;
        outp[(size_t)r * NCOLS + n] = acc[j] + bv;
    }
}

// ---------------------------------------------------------------------------
// TDM: DMA one 16x64-float K tile (rows strided by 512 floats) into LDS.
// D# group0: count=1 | lds_addr | global_addr(57b) | type=2  (ISA 08 §8.3)
// D# group1: data_size=4B, tensor_dim0=64, tensor_dim1=16, tile 64x16,
//            tensor_dim0_stride=512                          (ISA 08 §8.4)
// This toolchain exposes the 6-arg builtin:
//   (u32x4 g0, i32x8 g1, i32x4 g2, i32x4 g3, i32x8 extra, i32 cpol)
// ---------------------------------------------------------------------------
#if USE_TDM
__device__ __forceinline__ void tdm_load_tile(const float* gbase, unsigned lds_off)
{
    const unsigned long long ga = (unsigned long long)(uintptr_t)gbase;
    u32x4 g0;
    g0.x = 1u;                                            // count=1, user D#
    g0.y = lds_off;                                       // LDS byte address
    g0.z = (unsigned)(ga & 0xffffffffu);                  // global_addr[31:0]
    g0.w = (unsigned)((ga >> 32) & 0x01ffffffu) | (2u << 30); // [56:32] | type=2
    i32x8 g1;
    g1[0] = (2 << 16);               // workgroup_mask=0, data_size=2 (4 bytes)
    g1[1] = (int)(64u << 16);        // tensor_dim0[15:0]=64 (bits 63:48)
    g1[2] = (int)(16u << 16);        // tensor_dim0 hi=0 | tensor_dim1[15:0]=16
    g1[3] = (int)(64u << 16);        // tensor_dim1 hi=0 | tile_dim0=64
    g1[4] = 16;                      // tile_dim1=16, tile_dim2=0
    g1[5] = 512;                     // tensor_dim0_stride=512 elements
    g1[6] = 0;                       // stride hi | tensor_dim1_stride lo
    g1[7] = 0;
    const i32x4 z4 = {0, 0, 0, 0};            // groups 2/3 unused (2-D tensor)
    const i32x8 z8 = {0, 0, 0, 0, 0, 0, 0, 0};
    __builtin_amdgcn_tensor_load_to_lds(g0, g1, z4, z4, z8, 0);
}
#endif

// ---------------------------------------------------------------------------
// Fused attention: scores^T = K_tile x Q_tile^T (keys on M, queries on N),
// Gaussian prior from LDS table, online softmax, distance expectation.
// Block = 256 threads (8 waves) sharing K tiles staged in LDS (TDM, double
// buffered). Wave w handles queries [qbase0+16w, +16); each lane owns one
// query column; lanes L and L+16 split the 16 key rows.
// ---------------------------------------------------------------------------
__global__ __launch_bounds__(256)
void attn_kernel(const float* __restrict__ qbuf, const float* __restrict__ kbuf,
                 const float* __restrict__ prior_mean,
                 const float* __restrict__ log_prior_std,
                 float* __restrict__ out)
{
    __shared__ float tbl[TBL_N];
    __shared__ float ktile[2][16 * KD];   // double-buffered 4 KB K tiles

    const int group  = blockIdx.x & 15;   // 16 groups of 8 query tiles
    const int bh     = blockIdx.x >> 4;
    const int b      = bh >> 3;
    const int h      = bh & 7;
    const int qbase0 = group * 128;

    const float mean    = prior_mean[h];
    const float stdv    = __expf(log_prior_std[h]);
    const float coef    = 1.0f / (stdv * 2.5066282f);
    const float inv2var = 0.5f / (stdv * stdv);

    // prior table over dist = m - q for all (m, q) this block touches
    for (int i = threadIdx.x; i < TBL_N; i += 256) {
        const float d = (float)(i - (qbase0 + 127));
        const float t = d - mean;
        tbl[i] = coef * __expf(-t * t * inv2var);
    }

    const int wave = threadIdx.x >> 5;
    const int lane = threadIdx.x & 31;
    const int half = lane >> 4;
    const int ln   = lane & 15;

    const int   qbase = qbase0 + wave * 16;
    const int   qidx  = qbase + ln;
    const float qpos  = (float)qidx;

    const float* qrow  = qbuf + ((size_t)b * L_SEQ + qidx) * NCOLS + h * KD + 2 * half;
    const float* kbase = kbuf + ((size_t)b * L_SEQ) * NCOLS + h * KD;

    // Q fragments (B operand) live in registers for all 128 key tiles
    v2f bfrag[16];
#pragma unroll
    for (int kk = 0; kk < 16; ++kk)
        bfrag[kk] = *(const v2f*)(qrow + kk * 4);

#if USE_TDM
    if (wave == 0)
        tdm_load_tile(kbase, (unsigned)(uintptr_t)&ktile[0][0]);
#else
    {   // cooperative staging fallback: 256 threads x 16B
        const int row = threadIdx.x >> 4, cg = threadIdx.x & 15;
        *(float4*)&ktile[0][row * KD + cg * 4] =
            *(const float4*)(kbase + (size_t)row * NCOLS + cg * 4);
    }
#endif

    float mrun = -1e30f, Z = 0.0f, Nacc = 0.0f;
    const int tb = qbase0 + 127 - qidx;   // table index bias

    for (int mt = 0; mt < NT; ++mt) {
        const int cur = mt & 1;
#if USE_TDM
        if (wave == 0)
            __builtin_amdgcn_s_wait_tensorcnt(0);   // tile mt landed in LDS
#endif
        __syncthreads();                            // publish tile to all waves

        if (mt + 1 < NT) {                          // overlap DMA of mt+1
            const float* gnext = kbase + (size_t)(mt + 1) * 16 * NCOLS;
#if USE_TDM
            if (wave == 0)
                tdm_load_tile(gnext, (unsigned)(uintptr_t)&ktile[cur ^ 1][0]);
#else
            const int row = threadIdx.x >> 4, cg = threadIdx.x & 15;
            *(float4*)&ktile[cur ^ 1][row * KD + cg * 4] =
                *(const float4*)(gnext + (size_t)row * NCOLS + cg * 4);
#endif
        }

        // A fragments from LDS, then 16 chained f32 WMMAs over Kd=64
        const float* kl = &ktile[cur][ln * KD + 2 * half];
        v8f acc = {};
#pragma unroll
        for (int kk = 0; kk < 16; ++kk) {
            v2f a = *(const v2f*)(kl + kk * 4);
            acc = __builtin_amdgcn_wmma_f32_16x16x4_f32(false, a, false, bfrag[kk],
                                                        (short)0, acc, false, false);
        }

        const int mbase = mt * 16;
        const int mk0   = mbase + 8 * half;   // this lane's first key row
        float s2[8], tmax = -1e30f;
#pragma unroll
        for (int j = 0; j < 8; ++j) {
            const float p = tbl[mk0 + j + tb];            // Gaussian prior
            s2[j] = acc[j] * 0.125f * p;                  // * Kd^-0.5
            tmax  = fmaxf(tmax, s2[j]);
        }
        const float newm = fmaxf(mrun, tmax);
        const float corr = __expf(mrun - newm);
        Z *= corr;  Nacc *= corr;
#pragma unroll
        for (int j = 0; j < 8; ++j) {
            const float e = __expf(s2[j] - newm);
            Z    += e;
            Nacc += e * ((float)(mk0 + j) - qpos);        // sum attn * dist
        }
        mrun = newm;

        __syncthreads();   // all waves done reading ktile[cur] before reuse
    }

    // merge the two half-lanes that share a query column
    const float m2 = __shfl_xor(mrun, 16, 32);
    const float Z2 = __shfl_xor(Z,    16, 32);
    const float N2 = __shfl_xor(Nacc, 16, 32);
    const float M  = fmaxf(mrun, m2);
    const float c1 = __expf(mrun - M), c2 = __expf(m2 - M);
    const float Zt = Z * c1 + Z2 * c2;
    const float Nt = Nacc * c1 + N2 * c2;

    if (half == 0)
        out[((size_t)b * L_SEQ + qidx) * H_HEADS + h] = Nt / Zt;   // [B, L, H]
}

// ---------------------------------------------------------------------------
extern "C" void kernel_launch(void* const* d_in, const int* in_sizes, int n_in,
                              void* d_out, int out_size, void* d_ws, size_t ws_size,
                              hipStream_t stream)
{
    const float* x  = (const float*)d_in[0];
    const float* Wq = (const float*)d_in[1];
    const float* bq = (const float*)d_in[2];
    const float* Wk = (const float*)d_in[3];
    const float* bk = (const float*)d_in[4];
    const float* pm = (const float*)d_in[5];
    const float* ls = (const float*)d_in[6];
    float* out = (float*)d_out;

    float* qws = (float*)d_ws;                                    // 16 MB
    float* kws = qws + (size_t)B_BATCH * L_SEQ * NCOLS;           // 16 MB

    dim3 gp(128, 32, 2);      // 512 row tiles (4 waves/block), 32 col tiles, q/k
    proj_kernel<<<gp, 128, 0, stream>>>(x, Wq, bq, Wk, bk, qws, kws);

    dim3 ga(B_BATCH * H_HEADS * 16);   // 512 blocks x 8 waves = all query tiles
    attn_kernel<<<ga, 256, 0, stream>>>(qws, kws, pm, ls, out);
}